// FAGCNLayer_68143951118603
// MI455X (gfx1250) — compile-verified
//
#include <hip/hip_runtime.h>
#include <hip/hip_bf16.h>
#include <math.h>

#define HIDDEN 128

typedef __attribute__((ext_vector_type(2))) float v2f;
typedef __attribute__((ext_vector_type(8))) float v8f;

// ---------------------------------------------------------------------------
// Kernel 1: per-node dot products s1[n] = x[n]·w1, s2[n] = x[n]·w2
// One wave32 per 16-node tile, V_WMMA_F32_16X16X4_F32 accumulating over K=128.
//   A (16x4 f32): lane L holds row M = L%16, VGPR v holds K = v + 2*(L/16)
//   B (4x16 f32): lane L holds col N = L%16, VGPR v holds K = v + 2*(L/16)
//   C (16x16 f32): lane L holds col N = L%16, VGPR v holds row M = v + 8*(L/16)
// B column 0 = w1 chunk, column 1 = w2 chunk, other columns zero.
// ---------------------------------------------------------------------------
__global__ void node_dots_wmma(const float* __restrict__ x,
                               const float* __restrict__ att_w,
                               float* __restrict__ s1,
                               float* __restrict__ s2,
                               int n_nodes) {
    const int lane = threadIdx.x & 31;
    const int wave = threadIdx.x >> 5;
    const int tile = blockIdx.x * (blockDim.x >> 5) + wave;
    const int node_base = tile * 16;
    if (node_base >= n_nodes) return;          // wave-uniform exit, EXEC stays full

    const int n = lane & 15;                   // col index (A: row index)
    const int g = lane >> 4;                   // half-wave group

    int m = node_base + n;                     // node for this lane's A row
    if (m >= n_nodes) m = n_nodes - 1;         // clamp tail (dup rows, stores guarded)
    const float* __restrict__ arow = x + (size_t)m * HIDDEN + 2 * g;

    // B source: lane n==0 streams w1, n==1 streams w2, others masked to zero.
    const float* __restrict__ wsel = att_w + (n == 1 ? HIDDEN : 0) + 2 * g;
    const float bmask = (n < 2) ? 1.0f : 0.0f;

    v8f c = {0.f, 0.f, 0.f, 0.f, 0.f, 0.f, 0.f, 0.f};

#pragma unroll
    for (int k0 = 0; k0 < HIDDEN; k0 += 4) {
        v2f a = *(const v2f*)(arow + k0);
        v2f b = *(const v2f*)(wsel + k0);
        b = b * bmask;                          // branchless: EXEC stays all-ones
        // 8 args: (neg_a, A, neg_b, B, c_mod, C, reuse_a, reuse_b)
        c = __builtin_amdgcn_wmma_f32_16x16x4_f32(
                false, a, false, b, (short)0, c, false, false);
    }

    // C[m][0] -> s1, C[m][1] -> s2
    const int mo = node_base + 8 * g;
    if (n == 0) {
#pragma unroll
        for (int v = 0; v < 8; ++v)
            if (mo + v < n_nodes) s1[mo + v] = c[v];
    } else if (n == 1) {
#pragma unroll
        for (int v = 0; v < 8; ++v)
            if (mo + v < n_nodes) s2[mo + v] = c[v];
    }
}

// ---------------------------------------------------------------------------
// Kernel 2: per-edge scaled attention  alpha[e] = (1-eps)*tanh(s1[row]+s2[col]+b)
// ---------------------------------------------------------------------------
__global__ void edge_alpha_kernel(const int* __restrict__ ei,
                                  const float* __restrict__ s1,
                                  const float* __restrict__ s2,
                                  const float* __restrict__ att_b,
                                  const float* __restrict__ epsp,
                                  float* __restrict__ alpha,
                                  int n_edges) {
    const int e = blockIdx.x * blockDim.x + threadIdx.x;
    if (e >= n_edges) return;
    const int r = ei[e];
    const int cidx = ei[n_edges + e];
    const float logit = s1[r] + s2[cidx] + att_b[0];
    alpha[e] = (1.0f - epsp[0]) * tanhf(logit);
}

// ---------------------------------------------------------------------------
// Kernel 3: out = eps * x   (float4; also clears the harness poison each call)
// ---------------------------------------------------------------------------
__global__ void init_out_kernel(const float4* __restrict__ x4,
                                const float* __restrict__ epsp,
                                float4* __restrict__ out4,
                                size_t n4) {
    const size_t i = (size_t)blockIdx.x * blockDim.x + threadIdx.x;
    if (i >= n4) return;
    const float e = epsp[0];
    float4 v = x4[i];
    v.x *= e; v.y *= e; v.z *= e; v.w *= e;
    out4[i] = v;
}

// ---------------------------------------------------------------------------
// Kernel 4: wave32 per edge; gather x[row] (float4/lane, fully coalesced 512B),
// scale by alpha[e], scatter-add into out[col] with f32 atomics (L2-resident).
// ---------------------------------------------------------------------------
__global__ void edge_scatter_kernel(const int* __restrict__ ei,
                                    const float* __restrict__ x,
                                    const float* __restrict__ alpha,
                                    float* __restrict__ out,
                                    int n_edges) {
    const int lane = threadIdx.x & 31;
    const int wave = threadIdx.x >> 5;
    const int e = blockIdx.x * (blockDim.x >> 5) + wave;
    if (e >= n_edges) return;

    const int r = ei[e];
    const int cidx = ei[n_edges + e];
    const float a = alpha[e];

    const float4 xi = ((const float4*)(x + (size_t)r * HIDDEN))[lane];
    float* __restrict__ op = out + (size_t)cidx * HIDDEN + lane * 4;
    atomicAdd(op + 0, a * xi.x);
    atomicAdd(op + 1, a * xi.y);
    atomicAdd(op + 2, a * xi.z);
    atomicAdd(op + 3, a * xi.w);
}

// ---------------------------------------------------------------------------
// Launch
// inputs: 0:x [N,128] f32   1:edge_index [2,E] int   2:att_w [1,256] f32
//         3:att_b [1] f32   4:eps [1] f32
// ---------------------------------------------------------------------------
extern "C" void kernel_launch(void* const* d_in, const int* in_sizes, int n_in,
                              void* d_out, int out_size, void* d_ws, size_t ws_size,
                              hipStream_t stream) {
    const float* x     = (const float*)d_in[0];
    const int*   ei    = (const int*)d_in[1];
    const float* att_w = (const float*)d_in[2];
    const float* att_b = (const float*)d_in[3];
    const float* eps   = (const float*)d_in[4];
    float*       out   = (float*)d_out;

    const int n_nodes = in_sizes[0] / HIDDEN;
    const int n_edges = in_sizes[1] / 2;

    // workspace: s1[n_nodes] | s2[n_nodes] | alpha[n_edges]
    float* s1    = (float*)d_ws;
    float* s2    = s1 + n_nodes;
    float* alpha = s2 + n_nodes;

    // 1) node dot products via WMMA: 16 nodes per wave, 8 waves per block
    {
        const int tiles = (n_nodes + 15) / 16;
        const int waves_per_block = 8;
        const int blocks = (tiles + waves_per_block - 1) / waves_per_block;
        node_dots_wmma<<<blocks, waves_per_block * 32, 0, stream>>>(
            x, att_w, s1, s2, n_nodes);
    }

    // 2) per-edge alpha
    {
        const int threads = 256;
        const int blocks = (n_edges + threads - 1) / threads;
        edge_alpha_kernel<<<blocks, threads, 0, stream>>>(
            ei, s1, s2, att_b, eps, alpha, n_edges);
    }

    // 3) out = eps * x
    {
        const size_t n4 = (size_t)n_nodes * (HIDDEN / 4);
        const int threads = 256;
        const int blocks = (int)((n4 + threads - 1) / threads);
        init_out_kernel<<<blocks, threads, 0, stream>>>(
            (const float4*)x, eps, (float4*)out, n4);
    }

    // 4) scatter: one wave per edge
    {
        const int waves_per_block = 8;
        const int blocks = (n_edges + waves_per_block - 1) / waves_per_block;
        edge_scatter_kernel<<<blocks, waves_per_block * 32, 0, stream>>>(
            ei, x, alpha, out, n_edges);
    }
}